// MultiImage_TriFuse_Sampler_89481348644942
// MI455X (gfx1250) — compile-verified
//
#include <hip/hip_runtime.h>
#include <cmath>

typedef __attribute__((ext_vector_type(2))) float v2f;
typedef __attribute__((ext_vector_type(8))) float v8f;

#define NTRI  12288
#define NIMG  5
#define LN1E4 9.210340371976184f   // ln(10000)

static __device__ inline v8f wmma4(v2f a, v2f b, v8f c) {
  // D = A(16x4,f32) * B(4x16,f32) + C(16x16,f32)
  return __builtin_amdgcn_wmma_f32_16x16x4_f32(false, a, false, b, (short)0, c, false, false);
}
static __device__ inline v8f vzero8() { v8f z = {0.f,0.f,0.f,0.f,0.f,0.f,0.f,0.f}; return z; }

// ---------------------------------------------------------------------------
// K0: fold out_w + conv_w into W2[64][128], b2[64]
// ---------------------------------------------------------------------------
__global__ void k0_fold(const float* __restrict__ conv_w, const float* __restrict__ out_w,
                        const float* __restrict__ out_b, const float* __restrict__ conv_b,
                        float* __restrict__ W2, float* __restrict__ b2) {
  int i = blockIdx.x * 256 + threadIdx.x;
  if (i < 64 * 128) {
    int oc = i >> 7, c = i & 127;
    float s = 0.f;
    for (int e = 0; e < 128; ++e) s += conv_w[oc * 128 + e] * out_w[e * 128 + c];
    W2[i] = s;
  }
  if (i < 64) {
    float s = conv_b[i];
    for (int e = 0; e < 128; ++e) s += conv_w[i * 128 + e] * out_b[e];
    b2[i] = s;
  }
}

// ---------------------------------------------------------------------------
// K1: vol[b][j][y][x][c] (c<128: kimg, c>=128: vimg), each + bias + pos_enc(p)
// rows M = (b*5+j)*256 + p  (5120 rows), N = 256, K = 1280
// grid: 320 blocks x 256 threads; block = 16 rows x 256 cols (wave: 2 col-tiles)
// ---------------------------------------------------------------------------
__global__ __launch_bounds__(256)
void k1_kv(const float* __restrict__ img,            // [B][5][257][1280]
           const float* __restrict__ k_w, const float* __restrict__ k_b,
           const float* __restrict__ v_w, const float* __restrict__ v_b,
           float* __restrict__ vol)                  // [5120][256]
{
  __shared__ float As[16 * 128];
  const int tid  = threadIdx.x;
  const int lane = tid & 31, wave = tid >> 5;
  const int half = lane >> 4, lid = lane & 15;
  const int row0 = blockIdx.x * 16;

  v8f acc[2];
  acc[0] = vzero8(); acc[1] = vzero8();

  for (int kc = 0; kc < 1280; kc += 128) {
    __syncthreads();
    for (int t = tid; t < 16 * 128; t += 256) {
      int m = t >> 7, k = t & 127;
      int rowg = row0 + m;
      int bj = rowg >> 8;          // b*5+j
      int p  = rowg & 255;
      As[t] = img[((size_t)bj * 257 + 1 + p) * 1280 + kc + k];
    }
    __syncthreads();
#pragma unroll
    for (int ct = 0; ct < 2; ++ct) {
      int n = (wave + ct * 8) * 16 + lid;
      const float* Wr = (n < 128) ? (k_w + (size_t)n * 1280) : (v_w + (size_t)(n - 128) * 1280);
      for (int k0 = 0; k0 < 128; k0 += 4) {
        v2f a, bf;
        a.x = As[lid * 128 + k0 + 2 * half];
        a.y = As[lid * 128 + k0 + 2 * half + 1];
        bf.x = Wr[kc + k0 + 2 * half];
        bf.y = Wr[kc + k0 + 2 * half + 1];
        acc[ct] = wmma4(a, bf, acc[ct]);
      }
    }
  }

#pragma unroll
  for (int ct = 0; ct < 2; ++ct) {
    int n  = (wave + ct * 8) * 16 + lid;
    int c2 = n & 127;
    float bias = (n < 128) ? k_b[n] : v_b[n - 128];
    float div  = __expf(-LN1E4 * (float)(c2 & ~1) / 128.0f);
#pragma unroll
    for (int r = 0; r < 8; ++r) {
      int m = r + 8 * half;
      int rowg = row0 + m;
      int p = rowg & 255;
      float arg = (float)p * div;
      float pe  = (c2 & 1) ? cosf(arg) : sinf(arg);
      vol[(size_t)rowg * 256 + n] = acc[ct][r] + bias + pe;
    }
  }
}

// ---------------------------------------------------------------------------
// K2: q = tri^T @ q_w^T + q_b + pe_tri ; qp = q @ Wq^T + bq   -> qp[B][NTRI][128]
// grid: B*768 blocks (16 points each) x 256 threads; wave w = col-tile w
// ---------------------------------------------------------------------------
__global__ __launch_bounds__(256)
void k2_qproj(const float* __restrict__ tri,     // [B][64][NTRI]
              const float* __restrict__ q_w,     // [128][64]
              const float* __restrict__ q_b,     // [128]
              const float* __restrict__ in_w,    // [384][128] (rows 0..127 = Wq)
              const float* __restrict__ in_b,    // [384]
              float* __restrict__ qp)            // [B][NTRI][128]
{
  __shared__ float s_q[16 * 128];
  const int tid  = threadIdx.x;
  const int lane = tid & 31, wave = tid >> 5;
  const int half = lane >> 4, lid = lane & 15;
  const int b  = blockIdx.x / (NTRI / 16);
  const int n0 = (blockIdx.x % (NTRI / 16)) * 16;

  // ---- q tile (K = 64) ----
  v8f acc = vzero8();
  for (int k0 = 0; k0 < 64; k0 += 4) {
    int k = k0 + 2 * half;
    v2f a, bf;
    a.x = tri[((size_t)(b * 64 + k)) * NTRI + n0 + lid];
    a.y = tri[((size_t)(b * 64 + k + 1)) * NTRI + n0 + lid];
    const float* Wr = q_w + (size_t)(wave * 16 + lid) * 64 + k;
    bf.x = Wr[0]; bf.y = Wr[1];
    acc = wmma4(a, bf, acc);
  }
  {
    int c = wave * 16 + lid;
    float qb  = q_b[c];
    float div = __expf(-LN1E4 * (float)(c & ~1) / 128.0f);
#pragma unroll
    for (int r = 0; r < 8; ++r) {
      int m = r + 8 * half;
      float arg = (float)(n0 + m) * div;
      float pe  = (c & 1) ? cosf(arg) : sinf(arg);
      s_q[m * 128 + c] = acc[r] + qb + pe;
    }
  }
  __syncthreads();

  // ---- qp tile (K = 128) ----
  v8f acc2 = vzero8();
  for (int k0 = 0; k0 < 128; k0 += 4) {
    v2f a, bf;
    a.x = s_q[lid * 128 + k0 + 2 * half];
    a.y = s_q[lid * 128 + k0 + 2 * half + 1];
    const float* Wr = in_w + (size_t)(wave * 16 + lid) * 128 + k0 + 2 * half;
    bf.x = Wr[0]; bf.y = Wr[1];
    acc2 = wmma4(a, bf, acc2);
  }
  float bq = in_b[wave * 16 + lid];
#pragma unroll
  for (int r = 0; r < 8; ++r) {
    int m = r + 8 * half;
    qp[((size_t)(b * NTRI + n0 + m)) * 128 + wave * 16 + lid] = acc2[r] + bq;
  }
}

// ---------------------------------------------------------------------------
// K3: fused projection/sampling/attention/output.
// Block = 16 triplane points of batch b. 8 waves; wave w owns head w (dh=16).
// ---------------------------------------------------------------------------
__global__ __launch_bounds__(256)
void k3_attn(const float* __restrict__ vol,     // [B][5][16][16][256]
             const float* __restrict__ qp,      // [B][NTRI][128]
             const float* __restrict__ proj,    // [B][5][4][4]
             const float* __restrict__ in_w,    // [384][128]
             const float* __restrict__ in_b,    // [384]
             const float* __restrict__ W2,      // [64][128]
             const float* __restrict__ b2,      // [64]
             float* __restrict__ out)           // [B][64][NTRI]
{
  __shared__ float s_feat[16 * 128];
  __shared__ float s_qp[16 * 128];
  __shared__ float s_o[16 * 128];
  __shared__ float s_sc[NIMG][8][16];
  __shared__ int   s_ix0[NIMG][16], s_ix1[NIMG][16], s_iy0[NIMG][16], s_iy1[NIMG][16];
  __shared__ float s_w[NIMG][16][4];

  const int tid  = threadIdx.x;
  const int lane = tid & 31, wave = tid >> 5;
  const int half = lane >> 4, lid = lane & 15;
  const int b  = blockIdx.x / (NTRI / 16);
  const int n0 = (blockIdx.x % (NTRI / 16)) * 16;

  // ---- per-(point,image) bilinear coords ----
  if (tid < 16 * NIMG) {
    int m = tid & 15, j = tid >> 4;
    int n = n0 + m;
    int plane = n >> 12;
    int rr = (n >> 6) & 63;
    int cc = n & 63;
    float a  = (float)cc / 63.0f;
    float bb = (float)rr / 63.0f;
    const float scl = 2.0f * (1.0f + 0.1f + 1e-05f);
    float cx, cy, cz;
    if (plane == 0)      { cx = a;    cy = 0.5f; cz = bb;   }
    else if (plane == 1) { cx = a;    cy = bb;   cz = 0.5f; }
    else                 { cx = 0.5f; cy = a;    cz = bb;   }
    cx = (cx - 0.5f) * scl; cy = (cy - 0.5f) * scl; cz = (cz - 0.5f) * scl;
    const float* P = proj + (size_t)(b * NIMG + j) * 16;
    float c0 = P[0] * cx + P[1] * cy + P[2]  * cz + P[3];
    float c1 = P[4] * cx + P[5] * cy + P[6]  * cz + P[7];
    float c2 = P[8] * cx + P[9] * cy + P[10] * cz + P[11];
    float x = c0 / c2 / 223.0f;
    float y = c1 / c2 / 223.0f;
    float gx = fminf(1.f, fmaxf(-1.f, (x - 0.5f) * 2.0f));
    float gy = fminf(1.f, fmaxf(-1.f, (y - 0.5f) * 2.0f));
    float ix = (gx + 1.0f) * 0.5f * 15.0f;
    float iy = (gy + 1.0f) * 0.5f * 15.0f;
    float x0 = floorf(ix), y0 = floorf(iy);
    float wx = ix - x0,    wy = iy - y0;
    int x0i = min(15, max(0, (int)x0)); int x1i = min(15, x0i + 1);
    int y0i = min(15, max(0, (int)y0)); int y1i = min(15, y0i + 1);
    s_ix0[j][m] = x0i; s_ix1[j][m] = x1i; s_iy0[j][m] = y0i; s_iy1[j][m] = y1i;
    s_w[j][m][0] = (1.f - wy) * (1.f - wx);
    s_w[j][m][1] = (1.f - wy) * wx;
    s_w[j][m][2] = wy * (1.f - wx);
    s_w[j][m][3] = wy * wx;
  }

  // ---- stage qp tile ----
  for (int t = tid; t < 16 * 128; t += 256) {
    int m = t >> 7, c = t & 127;
    s_qp[t] = qp[((size_t)(b * NTRI + n0 + m)) * 128 + c];
  }
  __syncthreads();

  float qpr[8];
#pragma unroll
  for (int r = 0; r < 8; ++r) qpr[r] = s_qp[(r + 8 * half) * 128 + wave * 16 + lid];
  const float bk = in_b[128 + wave * 16 + lid];
  const float bv = in_b[256 + wave * 16 + lid];
  const float* Wk = in_w + 128 * 128;
  const float* Wv = in_w + 256 * 128;

  v8f vpf[NIMG];

#pragma unroll
  for (int j = 0; j < NIMG; ++j) {
    const float* volb = vol + (size_t)(b * NIMG + j) * 16 * 16 * 256;
    // sample K channels
    for (int t = tid; t < 16 * 128; t += 256) {
      int m = t >> 7, c = t & 127;
      int x0i = s_ix0[j][m], x1i = s_ix1[j][m], y0i = s_iy0[j][m], y1i = s_iy1[j][m];
      const float* w = s_w[j][m];
      s_feat[t] = w[0] * volb[(y0i * 16 + x0i) * 256 + c]
                + w[1] * volb[(y0i * 16 + x1i) * 256 + c]
                + w[2] * volb[(y1i * 16 + x0i) * 256 + c]
                + w[3] * volb[(y1i * 16 + x1i) * 256 + c];
    }
    __syncthreads();
    // kp tile for head `wave`
    v8f acck = vzero8();
    for (int k0 = 0; k0 < 128; k0 += 4) {
      v2f a, bf;
      a.x = s_feat[lid * 128 + k0 + 2 * half];
      a.y = s_feat[lid * 128 + k0 + 2 * half + 1];
      const float* Wr = Wk + (size_t)(wave * 16 + lid) * 128 + k0 + 2 * half;
      bf.x = Wr[0]; bf.y = Wr[1];
      acck = wmma4(a, bf, acck);
    }
    // scores: per-row dot over the 16 lanes holding head dims
#pragma unroll
    for (int r = 0; r < 8; ++r) {
      float v = (acck[r] + bk) * qpr[r];
      for (int off = 1; off < 16; off <<= 1) v += __shfl_xor(v, off, 32);
      if (lid == 0) s_sc[j][wave][r + 8 * half] = v * 0.25f;   // 1/sqrt(16)
    }
    __syncthreads();
    // sample V channels
    for (int t = tid; t < 16 * 128; t += 256) {
      int m = t >> 7, c = (t & 127) + 128;
      int x0i = s_ix0[j][m], x1i = s_ix1[j][m], y0i = s_iy0[j][m], y1i = s_iy1[j][m];
      const float* w = s_w[j][m];
      s_feat[t] = w[0] * volb[(y0i * 16 + x0i) * 256 + c]
                + w[1] * volb[(y0i * 16 + x1i) * 256 + c]
                + w[2] * volb[(y1i * 16 + x0i) * 256 + c]
                + w[3] * volb[(y1i * 16 + x1i) * 256 + c];
    }
    __syncthreads();
    // vp tile -> keep in registers
    v8f accv = vzero8();
    for (int k0 = 0; k0 < 128; k0 += 4) {
      v2f a, bf;
      a.x = s_feat[lid * 128 + k0 + 2 * half];
      a.y = s_feat[lid * 128 + k0 + 2 * half + 1];
      const float* Wr = Wv + (size_t)(wave * 16 + lid) * 128 + k0 + 2 * half;
      bf.x = Wr[0]; bf.y = Wr[1];
      accv = wmma4(a, bf, accv);
    }
    vpf[j] = accv;
    __syncthreads();
  }

  // ---- softmax over the 5 images, per (point, head) ----
  if (half == 0) {
    float sv[NIMG];
    float mx = -1e30f;
#pragma unroll
    for (int j = 0; j < NIMG; ++j) { sv[j] = s_sc[j][wave][lid]; mx = fmaxf(mx, sv[j]); }
    float sum = 0.f;
#pragma unroll
    for (int j = 0; j < NIMG; ++j) { sv[j] = __expf(sv[j] - mx); sum += sv[j]; }
    float inv = 1.0f / sum;
#pragma unroll
    for (int j = 0; j < NIMG; ++j) s_sc[j][wave][lid] = sv[j] * inv;
  }
  __syncthreads();

  // ---- o = sum_j attn * (vp + bv), in fragment layout -> LDS ----
#pragma unroll
  for (int r = 0; r < 8; ++r) {
    int m = r + 8 * half;
    float o = 0.f;
#pragma unroll
    for (int j = 0; j < NIMG; ++j) o += s_sc[j][wave][m] * (vpf[j][r] + bv);
    s_o[m * 128 + wave * 16 + lid] = o;
  }
  __syncthreads();

  // ---- final fused (out_w then conv_w) projection: 16x64 = s_o @ W2^T + b2 ----
  if (wave < 4) {
    v8f acc = vzero8();
    for (int k0 = 0; k0 < 128; k0 += 4) {
      v2f a, bf;
      a.x = s_o[lid * 128 + k0 + 2 * half];
      a.y = s_o[lid * 128 + k0 + 2 * half + 1];
      const float* Wr = W2 + (size_t)(wave * 16 + lid) * 128 + k0 + 2 * half;
      bf.x = Wr[0]; bf.y = Wr[1];
      acc = wmma4(a, bf, acc);
    }
    float bias = b2[wave * 16 + lid];
#pragma unroll
    for (int r = 0; r < 8; ++r) {
      int m  = r + 8 * half;
      int oc = wave * 16 + lid;
      out[((size_t)(b * 64 + oc)) * NTRI + n0 + m] = acc[r] + bias;
    }
  }
}

// ---------------------------------------------------------------------------
extern "C" void kernel_launch(void* const* d_in, const int* in_sizes, int n_in,
                              void* d_out, int out_size, void* d_ws, size_t ws_size,
                              hipStream_t stream) {
  const float* triplane = (const float*)d_in[0];
  const float* img      = (const float*)d_in[1];
  const float* proj     = (const float*)d_in[2];
  // d_in[3] = valid_frames (all ones, unused by reference math)
  const float* k_w = (const float*)d_in[4];
  const float* k_b = (const float*)d_in[5];
  const float* q_w = (const float*)d_in[6];
  const float* q_b = (const float*)d_in[7];
  const float* v_w = (const float*)d_in[8];
  const float* v_b = (const float*)d_in[9];
  const float* in_w = (const float*)d_in[10];
  const float* in_b = (const float*)d_in[11];
  const float* out_w = (const float*)d_in[12];
  const float* out_b = (const float*)d_in[13];
  const float* conv_w = (const float*)d_in[14];
  const float* conv_b = (const float*)d_in[15];
  float* out = (float*)d_out;

  float* vol = (float*)d_ws;                 // 4*5*256*256   = 1,310,720 floats
  float* qp  = vol + 1310720;                // 4*12288*128   = 6,291,456 floats
  float* W2  = qp + 6291456;                 // 64*128        = 8,192 floats
  float* b2  = W2 + 8192;                    // 64 floats

  k0_fold<<<32, 256, 0, stream>>>(conv_w, out_w, out_b, conv_b, W2, b2);
  k1_kv<<<320, 256, 0, stream>>>(img, k_w, k_b, v_w, v_b, vol);
  k2_qproj<<<4 * (NTRI / 16), 256, 0, stream>>>(triplane, q_w, q_b, in_w, in_b, qp);
  k3_attn<<<4 * (NTRI / 16), 256, 0, stream>>>(vol, qp, proj, in_w, in_b, W2, b2, out);
}